// LayerEngine_50371376447921
// MI455X (gfx1250) — compile-verified
//
#include <hip/hip_runtime.h>
#include <stdint.h>

#define BATCH 8
#define NCIN  12          // input channels
#define NC    11          // channels used
#define HH    512
#define WW    512
#define WT    16          // columns per block
#define TSEG  16          // threads cooperating per column
#define HSEG  (HH / TSEG) // 32 rows per thread
#define NTHREADS (WT * TSEG) // 256

#if defined(__HIP_DEVICE_COMPILE__) && __has_builtin(__builtin_amdgcn_tensor_load_to_lds)
#define USE_TDM 1
#else
#define USE_TDM 0
#endif

static __device__ __forceinline__ uint32_t lds_off(const void* p) {
    // low 32 bits of a generic pointer into LDS == LDS byte offset
    return (uint32_t)(uintptr_t)p;
}

#if USE_TDM
typedef unsigned int v4u __attribute__((ext_vector_type(4)));
typedef int          v8i __attribute__((ext_vector_type(8)));
typedef int          v4i __attribute__((ext_vector_type(4)));

// One TDM descriptor: 2D tile (WT x HH floats) out of a (HH x WW) f32 tensor.
// D# packing per CDNA5 ISA 8.3/8.4 (group0: count/lds/global/type, group1:
// data_size, tensor dims, tile dims, dim0 stride).
static __device__ __forceinline__ void tdm_load_tile(const float* gsrc, uint32_t lds_byte) {
    uint64_t ga = (uint64_t)(uintptr_t)gsrc;
    v4u g0;
    g0.x = 1u;                                   // count=1 (valid), user mode
    g0.y = lds_byte;                             // lds_addr
    g0.z = (uint32_t)(ga & 0xFFFFFFFFu);         // global_addr[31:0]
    g0.w = ((uint32_t)(ga >> 32) & 0x01FFFFFFu)  // global_addr[56:32]
           | (2u << 30);                         // type = 2 ("image")
    v8i g1;
    g1[0] = (int)(2u << 16);                               // data_size=2 -> 4B elements
    g1[1] = (int)((WW & 0xFFFF) << 16);                    // tensor_dim0 lo16
    g1[2] = (int)((WW >> 16) | ((HH & 0xFFFF) << 16));     // tensor_dim0 hi16 | tensor_dim1 lo16
    g1[3] = (int)((HH >> 16) | (WT << 16));                // tensor_dim1 hi16 | tile_dim0=WT
    g1[4] = (int)HH;                                       // tile_dim1=HH, tile_dim2=0
    g1[5] = (int)WW;                                       // tensor_dim0_stride lo32
    g1[6] = 0;                                             // stride hi16 | dim1_stride lo16
    g1[7] = 0;
    v4i z4 = {0, 0, 0, 0};
    v8i z8 = {0, 0, 0, 0, 0, 0, 0, 0};
    // 6-arg form (clang-23 / therock-10.0 toolchain)
    __builtin_amdgcn_tensor_load_to_lds(g0, g1, z4, z4, z8, 0);
}
#endif

// Issue the (possibly asynchronous) global -> LDS tile copy.
static __device__ __forceinline__ void issue_tile_load(const float* src, float* dstLds, int tid) {
#if USE_TDM
    if (tid == 0) {                       // one DMA per workgroup; EXEC ignored for tensor ops
        tdm_load_tile(src, lds_off(dstLds));
    }
#else
    const int nf4 = HH * WT / 4;          // 2048 float4 packets
    for (int f = tid; f < nf4; f += NTHREADS) {
        int h  = f / (WT / 4);
        int c4 = f % (WT / 4);
        const float* g = src + (size_t)h * WW + c4 * 4;
        uint32_t l = lds_off(dstLds + h * WT + c4 * 4);
        asm volatile("global_load_async_to_lds_b128 %0, %1, off"
                     :: "v"(l), "v"(g) : "memory");
    }
#endif
}

// Wait for the older in-flight tile; `newer_pending` = a prefetch for the next
// tile is still allowed to be outstanding.
static __device__ __forceinline__ void wait_tile(bool newer_pending, int tid) {
#if USE_TDM
    if (tid == 0) {
        if (newer_pending) __builtin_amdgcn_s_wait_tensorcnt(1);
        else               __builtin_amdgcn_s_wait_tensorcnt(0);
    }
#else
    (void)tid;
    if (newer_pending) asm volatile("s_wait_asynccnt 0x8" ::: "memory"); // 8 issues per wave per tile
    else               asm volatile("s_wait_asynccnt 0x0" ::: "memory");
#endif
}

__global__ __launch_bounds__(NTHREADS)
void layer_stage1(const float* __restrict__ in,
                  float* __restrict__ lsm_out,
                  float* __restrict__ clean_out,
                  float* __restrict__ std_out,
                  float* __restrict__ lp_ws)
{
    __shared__ __align__(16) float tile2[2][HH * WT]; // 64 KB: double-buffered column tiles
    __shared__ float carry[HH * WT];                  // 32 KB: cross-channel scan state
    __shared__ float red0[TSEG * WT];
    __shared__ float red1[TSEG * WT];
    __shared__ float red2[TSEG * WT];
    __shared__ float colMax[WT], colLogZ[WT], colInvS[WT], colMean[WT];

    const int tid = threadIdx.x;
    const int tw  = tid % WT;        // column within tile
    const int th  = tid / WT;        // row-segment index
    const int b   = blockIdx.x / (WW / WT);
    const int w0  = (blockIdx.x % (WW / WT)) * WT;
    const int h0  = th * HSEG;

    const float* chan0 = in + (((size_t)b * NCIN) * HH) * WW + w0;
    const size_t chan_stride = (size_t)HH * WW;

    // Prime the pipeline with channel 0.
    issue_tile_load(chan0, tile2[0], tid);

    for (int c = 0; c < NC; ++c) {
        float* tileX = tile2[c & 1];

        // Prefetch next channel into the other buffer (safe: trailing barrier
        // of iteration c-1 guarantees nobody still reads that buffer).
        if (c + 1 < NC)
            issue_tile_load(chan0 + (size_t)(c + 1) * chan_stride, tile2[(c + 1) & 1], tid);

        wait_tile(c + 1 < NC, tid);
        __syncthreads();

        // ---- Phase B: column max ----
        float m = -3.402823466e38f;
        for (int i = 0; i < HSEG; ++i)
            m = fmaxf(m, tileX[(h0 + i) * WT + tw]);
        red0[th * WT + tw] = m;
        __syncthreads();
        if (th == 0) {
            float mm = red0[tw];
            for (int j = 1; j < TSEG; ++j) mm = fmaxf(mm, red0[j * WT + tw]);
            colMax[tw] = mm;
        }
        __syncthreads();
        m = colMax[tw];

        // ---- Phase C: sum(exp), sum(exp*h) ----
        float s0 = 0.f, s1 = 0.f;
        for (int i = 0; i < HSEG; ++i) {
            float e = expf(tileX[(h0 + i) * WT + tw] - m);
            s0 += e;
            s1 += e * (float)(h0 + i);
        }
        red0[th * WT + tw] = s0;   // kept for Phase D prefix
        red1[th * WT + tw] = s1;
        __syncthreads();
        if (th == 0) {
            float S0 = 0.f, S1 = 0.f;
            for (int j = 0; j < TSEG; ++j) { S0 += red0[j * WT + tw]; S1 += red1[j * WT + tw]; }
            colInvS[tw] = 1.0f / S0;
            colLogZ[tw] = logf(S0);
            colMean[tw] = S1 / S0;
        }
        __syncthreads();

        // ---- Phase C2: centered second moment -> std, layer position ----
        const float mean = colMean[tw];
        float s2 = 0.f;
        for (int i = 0; i < HSEG; ++i) {
            float e = expf(tileX[(h0 + i) * WT + tw] - m);
            float d = (float)(h0 + i) - mean;
            s2 += e * d * d;
        }
        red2[th * WT + tw] = s2;
        __syncthreads();
        if (th == 0) {
            float S2 = 0.f;
            for (int j = 0; j < TSEG; ++j) S2 += red2[j * WT + tw];
            float var = S2 * colInvS[tw];
            size_t o = ((size_t)b * NC + c) * WW + w0 + tw;
            std_out[o] = sqrtf(fmaxf(var, 0.0f));
            lp_ws[o]   = colMean[tw];
        }

        // ---- Phase D: log-softmax write + column cumsum + cross-channel scan ----
        float pre = 0.f;
        for (int j = 0; j < th; ++j) pre += red0[j * WT + tw]; // exclusive prefix
        const float invS = colInvS[tw];
        const float lz   = colLogZ[tw];
        float run = pre;
        float* lsm_base   = lsm_out   + (((size_t)b * NC   + c) * HH) * WW + w0 + tw;
        float* clean_base = clean_out + (((size_t)b * NCIN + c) * HH) * WW + w0 + tw;
        for (int i = 0; i < HSEG; ++i) {
            int h = h0 + i;
            float x = tileX[h * WT + tw];
            float e = expf(x - m);
            run += e;
            float cum = run * invS;
            __builtin_nontemporal_store(x - m - lz, &lsm_base[(size_t)h * WW]);
            float cl;
            if (c == 0) {
                cl = 1.0f - cum;               // clean[0] = 1 - cum_ch0
                carry[h * WT + tw] = cum;      // tc[1]
            } else {
                float cprev = carry[h * WT + tw];
                float tcn = fmaxf(cum + cprev - 1.0f, 0.0f);
                cl = cprev - tcn;
                carry[h * WT + tw] = tcn;
                if (c == NC - 1) {
                    __builtin_nontemporal_store(
                        tcn, &clean_out[(((size_t)b * NCIN + 11) * HH + h) * WW + w0 + tw]);
                }
            }
            __builtin_nontemporal_store(cl, &clean_base[(size_t)h * WW]);
        }
        __syncthreads(); // protect tile buffer / red arrays before next channel
    }
}

__global__ void layer_stage2(const float* __restrict__ lp,
                             float* __restrict__ clp_out,   // (8,11,512)
                             float* __restrict__ topo_out,  // (8,10,512)
                             float* __restrict__ cont_out,  // (8,11,511)
                             float* __restrict__ curv_out)  // (8,11,512)
{
    int idx = blockIdx.x * blockDim.x + threadIdx.x;
    if (idx >= BATCH * WW) return;
    int b = idx / WW, w = idx % WW;
    const float cmax[NC] = {1.2261f, 1.1558f, 1.1161f, 1.1195f, 2.7202f, 2.3714f,
                            1.7055f, 3.2717f, 2.6716f, 5.0418f, 0.4293f};
    int wm5 = (w - 5 < 0) ? 0 : (w - 5);
    int wp5 = (w + 5 > WW - 1) ? (WW - 1) : (w + 5);
    float runmax = -3.402823466e38f;
    float prev = 0.f;
    for (int c = 0; c < NC; ++c) {
        const float* base = lp + ((size_t)b * NC + c) * WW;
        float v = base[w];
        runmax = fmaxf(runmax, v);
        clp_out[((size_t)b * NC + c) * WW + w] = runmax;
        if (c > 0)
            topo_out[((size_t)b * 10 + (c - 1)) * WW + w] = fmaxf(prev - v, 0.f);
        if (w > 0) {
            float vm1 = base[w - 1];
            cont_out[((size_t)b * NC + c) * (WW - 1) + (w - 1)] = fabsf(vm1 - v);
        }
        float a  = base[wm5];
        float bb = base[wp5];
        float fo = bb - a;
        float so = a - 2.0f * v + bb;
        float den = 1.0f + fo * fo;
        float curv = so / (den * sqrtf(den));
        curv_out[((size_t)b * NC + c) * WW + w] = fabsf(curv) - cmax[c];
        prev = v;
    }
}

extern "C" void kernel_launch(void* const* d_in, const int* in_sizes, int n_in,
                              void* d_out, int out_size, void* d_ws, size_t ws_size,
                              hipStream_t stream) {
    (void)in_sizes; (void)n_in; (void)out_size; (void)ws_size;
    const float* in = (const float*)d_in[0];
    float* out = (float*)d_out;

    const size_t SZ_LSM   = (size_t)BATCH * NC   * HH * WW;  // 23,068,672
    const size_t SZ_CLP   = (size_t)BATCH * NC   * WW;       //     45,056
    const size_t SZ_CLEAN = (size_t)BATCH * NCIN * HH * WW;  // 25,165,824
    const size_t SZ_STD   = (size_t)BATCH * NC   * WW;
    const size_t SZ_TOPO  = (size_t)BATCH * 10   * WW;
    const size_t SZ_CONT  = (size_t)BATCH * NC   * (WW - 1);

    float* o_lsm   = out;
    float* o_clp   = o_lsm   + SZ_LSM;
    float* o_clean = o_clp   + SZ_CLP;
    float* o_std   = o_clean + SZ_CLEAN;
    float* o_topo  = o_std   + SZ_STD;
    float* o_cont  = o_topo  + SZ_TOPO;
    float* o_curv  = o_cont  + SZ_CONT;

    float* lp = (float*)d_ws; // 8*11*512 floats = 176 KB scratch

    dim3 grid1(BATCH * (WW / WT)); // 256 blocks
    layer_stage1<<<grid1, NTHREADS, 0, stream>>>(in, o_lsm, o_clean, o_std, lp);

    int n2 = BATCH * WW;
    layer_stage2<<<(n2 + 255) / 256, 256, 0, stream>>>(lp, o_clp, o_topo, o_cont, o_curv);
}